// MPLayerWithUpdate_352187319162
// MI455X (gfx1250) — compile-verified
//
#include <hip/hip_runtime.h>

typedef float v2f __attribute__((ext_vector_type(2)));
typedef float v8f __attribute__((ext_vector_type(8)));

#define N_NODES 100000
#define N_EDGES 1600000
#define CH 64

// ---------------------------------------------------------------------------
// Zero the scratch accumulator S[N_NODES][64] and deg[N_NODES]
// ---------------------------------------------------------------------------
__global__ __launch_bounds__(256) void zero_ws_kernel(float* __restrict__ ws, int n) {
    int i = blockIdx.x * 256 + threadIdx.x;
    if (i < n) ws[i] = 0.0f;
}

// ---------------------------------------------------------------------------
// Edge scatter: S[dst] += x[src]; deg[dst] += 1
// One wave32 per edge: 32 lanes x float2 = 64 channels. src/dst are
// wave-uniform, the 256B gather is one coalesced b64 load per lane, and the
// accumulator updates are hardware global_atomic_add_f32 (L2-resident).
// ---------------------------------------------------------------------------
__global__ __launch_bounds__(256) void scatter_edges_kernel(
    const float* __restrict__ x, const long long* __restrict__ ei,
    float* __restrict__ S, float* __restrict__ deg) {
    long long t = (long long)blockIdx.x * 256 + threadIdx.x;
    int e    = (int)(t >> 5);         // one wave per edge
    int lane = (int)(t & 31);
    if (e >= N_EDGES) return;
    long long src = ei[e];
    long long dst = ei[N_EDGES + e];
    const float* xs = x + src * CH + lane * 2;
    float*       sd = S + dst * CH + lane * 2;
    v2f v = *(const v2f*)xs;          // global_load_b64
    unsafeAtomicAdd(sd + 0, v.x);     // global_atomic_add_f32
    unsafeAtomicAdd(sd + 1, v.y);
    if (lane == 0) unsafeAtomicAdd(&deg[dst], 1.0f);
}

// ---------------------------------------------------------------------------
// out = diag(1/max(deg,1)) * (S @ W^T) + min(deg,1)*b + x
// One wave32 per 16x16 output tile using V_WMMA_F32_16X16X4_F32.
// Row-normalization commutes with the GEMM, so it is applied to the D rows
// in the epilogue -> WMMA operands come straight from global_load_b64.
//   A layout: lane l<16 -> row m0+l, VGPR{0,1} = K{0,1}; lane>=16 -> K{2,3}
//   B layout: lane l<16 -> col n0+l, VGPR{0,1} = K{0,1}; lane>=16 -> K{2,3}
//   D layout: VGPR r -> row m0 + 8*(lane>=16) + r, col n0 + (lane&15)
// ---------------------------------------------------------------------------
__global__ __launch_bounds__(256) void gemm_update_kernel(
    const float* __restrict__ S, const float* __restrict__ deg,
    const float* __restrict__ W, const float* __restrict__ b,
    const float* __restrict__ x, float* __restrict__ out) {
    int wave = (blockIdx.x * 256 + threadIdx.x) >> 5;
    int lane = threadIdx.x & 31;
    const int tilesN = CH / 16;                  // 4 column tiles
    int mt = wave / tilesN;
    int nt = wave % tilesN;
    if (mt >= N_NODES / 16) return;              // never taken: grid is exact
    int m0 = mt * 16, n0 = nt * 16;
    int half = lane >> 4;                        // 0: K=0,1  1: K=2,3
    int l = lane & 15;

    const float* Srow = S + (long)(m0 + l) * CH + 2 * half;
    const float* Wcol = W + (long)(n0 + l) * CH + 2 * half;   // B[k][n] = W[n][k]

    v8f acc = {};
    #pragma unroll
    for (int kk = 0; kk < CH; kk += 4) {
        v2f a  = *(const v2f*)(Srow + kk);       // 8B-aligned global_load_b64
        v2f bb = *(const v2f*)(Wcol + kk);
        acc = __builtin_amdgcn_wmma_f32_16x16x4_f32(
            /*neg_a=*/false, a, /*neg_b=*/false, bb,
            /*c_mod=*/(short)0, acc, /*reuse_a=*/false, /*reuse_b=*/false);
    }

    // Epilogue: per-row mean normalization (rcp + 1 Newton step), bias scaled
    // by min(deg,1) (zero-degree rows get exactly 0), residual add.
    int ocol = n0 + l;
    float bv = b[ocol];
    #pragma unroll
    for (int r = 0; r < 8; ++r) {
        int row    = m0 + half * 8 + r;
        float d    = deg[row];
        float dmax = fmaxf(d, 1.0f);
        float rinv = __builtin_amdgcn_rcpf(dmax);
        rinv       = rinv * (2.0f - dmax * rinv);      // ~0.5 ulp
        float bscale = fminf(d, 1.0f);
        long idx = (long)row * CH + ocol;
        out[idx] = x[idx] + acc[r] * rinv + bscale * bv;
    }
}

// ---------------------------------------------------------------------------
extern "C" void kernel_launch(void* const* d_in, const int* in_sizes, int n_in,
                              void* d_out, int out_size, void* d_ws, size_t ws_size,
                              hipStream_t stream) {
    const float*     x  = (const float*)d_in[0];
    const float*     W  = (const float*)d_in[1];
    const float*     b  = (const float*)d_in[2];
    const long long* ei = (const long long*)d_in[3];   // int64 edge_index [2,E]
    float* out = (float*)d_out;

    float* S   = (float*)d_ws;                         // [N_NODES, 64]
    float* deg = S + (size_t)N_NODES * CH;             // [N_NODES]

    int ztotal = N_NODES * CH + N_NODES;
    zero_ws_kernel<<<(ztotal + 255) / 256, 256, 0, stream>>>(S, ztotal);

    long long tthreads = (long long)N_EDGES * 32;      // one wave per edge
    int sblocks = (int)((tthreads + 255) / 256);
    scatter_edges_kernel<<<sblocks, 256, 0, stream>>>(x, ei, S, deg);

    int waves   = (N_NODES / 16) * (CH / 16);          // 25000 tiles
    int gblocks = (waves + 7) / 8;                     // 8 waves per 256-thread block
    gemm_update_kernel<<<gblocks, 256, 0, stream>>>(S, deg, W, b, x, out);
}